// Loss_17282948399165
// MI455X (gfx1250) — compile-verified
//
#include <hip/hip_runtime.h>
#include <hip/hip_bf16.h>
#include <stdint.h>

// ---------------- configuration ----------------
#define TPB            256
#define POS_PER_BLOCK  1024      // spatial positions per gram block (4 per thread)
#define KC             4         // channels per async chunk (double buffered)
#define CONTENT_BLOCKS 512
#define CONTENT_NF4    2097152   // 4*512*64*64 / 4
#define TOTAL_BLOCKS   3240

struct JobTable {
  const float* ptr[11];  // jobs 0-4 styles, 5-9 pics, 10 content0
  const float* pic3;     // second operand of content job
  int C[11];
  int HW[11];
  int lhw[11];           // log2(HW)
  int bs[12];            // block prefix per job
};

// ---------------- CDNA5 async copy helpers ----------------
__device__ __forceinline__ void async_ld_b128(uint32_t lds_byte_addr,
                                              uint32_t glb_byte_off,
                                              const float* base) {
  // GVS mode: copies 16 bytes from global saddr(64b base) + vaddr(32b byte offset) to the LDS address in vdst
  asm volatile("global_load_async_to_lds_b128 %0, %1, %2"
               :: "v"(lds_byte_addr), "v"(glb_byte_off), "s"(base)
               : "memory");
}
#define WAIT_ASYNCCNT(n) asm volatile("s_wait_asynccnt " #n ::: "memory")
#define WAIT_DSCNT0()    asm volatile("s_wait_dscnt 0"     ::: "memory")

// ---------------- fused reduction kernel ----------------
__global__ __launch_bounds__(TPB)
void style_loss_partials(JobTable jt, float* __restrict__ partials)
{
  __shared__ float ldsbuf[2 * KC * POS_PER_BLOCK];  // 32 KB staging
  __shared__ float red[TPB];

  const int tid = threadIdx.x;
  int blk = blockIdx.x;
  int j = 0;
  while (blk >= jt.bs[j + 1]) ++j;          // 11 jobs: tiny scalar search
  const int lblk = blk - jt.bs[j];

  float val = 0.0f;

  if (j < 10) {
    // ---- gram-sum job: sum_{pos} (sum_c x[b,c,pos])^2 ----
    const float* __restrict__ x = jt.ptr[j];
    const int C   = jt.C[j];
    const int HW  = jt.HW[j];
    const int lhw = jt.lhw[j];
    const uint32_t chs = (uint32_t)HW * 4u;            // channel stride, bytes

    const int pos = lblk * POS_PER_BLOCK + tid * 4;    // 4 consecutive positions
    const int b   = pos >> lhw;
    const int hw  = pos & (HW - 1);
    const uint32_t voff0 =
        ((uint32_t)(b * C) * (uint32_t)HW + (uint32_t)hw) * 4u;

    // per-lane LDS byte address (low 32 bits of generic ptr == LDS offset)
    const uint32_t lb =
        (uint32_t)(uintptr_t)(&ldsbuf[0]) + (uint32_t)tid * 16u;

    float a0 = 0.f, a1 = 0.f, a2 = 0.f, a3 = 0.f;
    const int nch = C / KC;

    // prime: chunk 0 -> buffer 0
#pragma unroll
    for (int k = 0; k < KC; ++k)
      async_ld_b128(lb + (uint32_t)k * 4096u, voff0 + (uint32_t)k * chs, x);

    for (int ch = 0; ch < nch; ++ch) {
      if (ch + 1 < nch) {
        // before overwriting buffer (ch+1)&1, make sure our DS reads of it
        // (from chunk ch-1) have completed: async LDS writes are unordered
        // w.r.t. DS loads.
        WAIT_DSCNT0();
        const uint32_t vo  = voff0 + (uint32_t)(ch + 1) * (uint32_t)KC * chs;
        const uint32_t lbn = lb + (uint32_t)(((ch + 1) & 1) * KC * 4096);
#pragma unroll
        for (int k = 0; k < KC; ++k)
          async_ld_b128(lbn + (uint32_t)k * 4096u, vo + (uint32_t)k * chs, x);
        WAIT_ASYNCCNT(4);   // chunk ch landed (loads complete in order)
      } else {
        WAIT_ASYNCCNT(0);   // last chunk landed
      }
      const int bufo = (ch & 1) * (KC * POS_PER_BLOCK);
#pragma unroll
      for (int k = 0; k < KC; ++k) {
        const float4 v =
            *(const float4*)&ldsbuf[bufo + k * POS_PER_BLOCK + tid * 4];
        a0 += v.x; a1 += v.y; a2 += v.z; a3 += v.w;
      }
    }
    // square the per-position channel sums
    val = a0 * a0 + a1 * a1 + a2 * a2 + a3 * a3;
  } else {
    // ---- content job: sum (content0 - pic3)^2 ----
    const float4* __restrict__ c = (const float4*)jt.ptr[10];
    const float4* __restrict__ p = (const float4*)jt.pic3;
    const int stride = CONTENT_BLOCKS * TPB;
    for (int i = lblk * TPB + tid; i < CONTENT_NF4; i += stride) {
      const float4 a = c[i];
      const float4 b4 = p[i];
      const float dx = a.x - b4.x, dy = a.y - b4.y,
                  dz = a.z - b4.z, dw = a.w - b4.w;
      val += dx * dx + dy * dy + dz * dz + dw * dw;
    }
  }

  // deterministic block tree-reduction
  red[tid] = val;
  __syncthreads();
#pragma unroll
  for (int s = TPB / 2; s > 0; s >>= 1) {
    if (tid < s) red[tid] += red[tid + s];
    __syncthreads();
  }
  if (tid == 0) partials[blockIdx.x] = red[0];
}

// ---------------- finalize: combine partials (double precision) ------------
__global__ void style_loss_finalize(const float* __restrict__ partials,
                                    float* __restrict__ out)
{
  __shared__ double s[11];
  const int bs[12] = {0, 1024, 1280, 1344, 1360, 1364,
                      2388, 2644, 2708, 2724, 2728, TOTAL_BLOCKS};
  const int t = threadIdx.x;
  if (t < 11) {
    double acc = 0.0;
    for (int i = bs[t]; i < bs[t + 1]; ++i) acc += (double)partials[i];
    s[t] = acc;
  }
  __syncthreads();
  if (t == 0) {
    const double Cv[5]  = {64.0, 128.0, 256.0, 512.0, 512.0};
    const double HWv[5] = {262144.0, 65536.0, 16384.0, 4096.0, 1024.0};
    double style = 0.0;
    for (int i = 0; i < 5; ++i) {
      const double g = s[5 + i];   // pic gram sum
      const double a = s[i];       // style gram sum
      const double d = g - a;
      style += 0.2 * (d * d) / (Cv[i] * Cv[i]) / (HWv[i] * HWv[i]) / 4.0;
    }
    const double content = 0.5 * s[10];
    *out = (float)(1.0 * content + 1000.0 * style);
  }
}

// ---------------- host entry ----------------
extern "C" void kernel_launch(void* const* d_in, const int* in_sizes, int n_in,
                              void* d_out, int out_size, void* d_ws, size_t ws_size,
                              hipStream_t stream) {
  (void)in_sizes; (void)n_in; (void)out_size; (void)ws_size;

  static const int Cs[5]  = {64, 128, 256, 512, 512};
  static const int HWs[5] = {262144, 65536, 16384, 4096, 1024};
  static const int LHW[5] = {18, 16, 14, 12, 10};

  JobTable jt;
  for (int i = 0; i < 5; ++i) {
    jt.ptr[i]     = (const float*)d_in[i];       // style_i
    jt.ptr[5 + i] = (const float*)d_in[5 + i];   // pic_i
    jt.C[i] = jt.C[5 + i] = Cs[i];
    jt.HW[i] = jt.HW[5 + i] = HWs[i];
    jt.lhw[i] = jt.lhw[5 + i] = LHW[i];
  }
  jt.ptr[10] = (const float*)d_in[10];           // content0
  jt.pic3    = (const float*)d_in[8];            // pic3
  jt.C[10] = 0; jt.HW[10] = 0; jt.lhw[10] = 0;

  const int bs[12] = {0, 1024, 1280, 1344, 1360, 1364,
                      2388, 2644, 2708, 2724, 2728, TOTAL_BLOCKS};
  for (int i = 0; i < 12; ++i) jt.bs[i] = bs[i];

  float* partials = (float*)d_ws;

  style_loss_partials<<<TOTAL_BLOCKS, TPB, 0, stream>>>(jt, partials);
  style_loss_finalize<<<1, 32, 0, stream>>>(partials, (float*)d_out);
}